// CapsuleLayer_38680475467933
// MI455X (gfx1250) — compile-verified
//
#include <hip/hip_runtime.h>
#include <math.h>

// CDNA5 wave32 WMMA fragment types
typedef float v2f __attribute__((ext_vector_type(2)));
typedef float v4f __attribute__((ext_vector_type(4)));
typedef float v8f __attribute__((ext_vector_type(8)));

#define NWAVES 8
#define THREADS (NWAVES * 32)
#define B_TOT 512
#define R_TOT 1152
#define RPW (R_TOT / NWAVES)   // 144 routes per wave
#define C_CAPS 10
#define O_DIM 16
#define I_DIM 8
#define MTILE 16               // batch tile per block
#define S_ELEMS (MTILE * C_CAPS * O_DIM)   // 2560
#define W_RSTRIDE (C_CAPS * O_DIM * I_DIM) // 1280 floats per route

// One routing phase. PHASE==0: uniform coupling (softmax of zeros = 1/C).
// PHASE==1: logits = u.v0, v-buffer updated to v0+v1.
// PHASE==2: logits = u.(v0+v1), writes final v2 to global.
template <int PHASE>
__device__ __forceinline__ void phase_body(
    const float* __restrict__ x, const float* __restrict__ W,
    const float* __restrict__ bias, float* __restrict__ out,
    float* s_lds, float* vdot_lds,
    int tid, int wave, int lane, int bl, int hh, int mb)
{
    // zero the s reduction buffer (barrier also orders prior-phase v writes)
    for (int i = tid; i < S_ELEMS; i += THREADS) s_lds[i] = 0.0f;
    __syncthreads();

    // s_j accumulated DIRECTLY in WMMA C operands across the r loop:
    // S_c[m=o, n=b] += A_W x (B_x * diag(cij[:,c]))
    v8f s_wacc[C_CAPS];
#pragma unroll
    for (int c = 0; c < C_CAPS; ++c) s_wacc[c] = (v8f){};

    // running pointers: strength-reduced induction over the route chunk
    const int r0 = wave * RPW;
    const float* xp = x + ((size_t)(mb + bl) * R_TOT + r0) * I_DIM + 2 * hh;
    const float* wp = W + (size_t)r0 * W_RSTRIDE + bl * I_DIM + 2 * hh;
    const float* vbase = &vdot_lds[(bl * C_CAPS) * O_DIM + 8 * hh];

    for (int rr = 0; rr < RPW; ++rr) {
        // B fragment (x tile): lane col n = bl, k = v + 2*hh (+4 per k-step)
        const v2f bfrag0 = *reinterpret_cast<const v2f*>(xp);
        const v2f bfrag1 = *reinterpret_cast<const v2f*>(xp + 4);

        if constexpr (PHASE == 0) {
            const v2f wb0 = bfrag0 * 0.1f;
            const v2f wb1 = bfrag1 * 0.1f;
#pragma unroll
            for (int c = 0; c < C_CAPS; ++c) {
                const float* wa = wp + c * (O_DIM * I_DIM);
                const v2f a0 = *reinterpret_cast<const v2f*>(wa);
                const v2f a1 = *reinterpret_cast<const v2f*>(wa + 4);
                s_wacc[c] = __builtin_amdgcn_wmma_f32_16x16x4_f32(
                        false, a0, false, wb0, (short)0, s_wacc[c], false, false);
                s_wacc[c] = __builtin_amdgcn_wmma_f32_16x16x4_f32(
                        false, a1, false, wb1, (short)0, s_wacc[c], false, false);
            }
        } else {
            // ---- agreement pass: logits[c] = u_hat . v (v read from LDS) ----
            float cij[C_CAPS];
#pragma unroll
            for (int c = 0; c < C_CAPS; ++c) {
                const float* wa = wp + c * (O_DIM * I_DIM);
                const v2f a0 = *reinterpret_cast<const v2f*>(wa);
                const v2f a1 = *reinterpret_cast<const v2f*>(wa + 4);
                v8f u = {};
                u = __builtin_amdgcn_wmma_f32_16x16x4_f32(
                        false, a0, false, bfrag0, (short)0, u, false, false);
                u = __builtin_amdgcn_wmma_f32_16x16x4_f32(
                        false, a1, false, bfrag1, (short)0, u, false, false);
                const v4f* vp = reinterpret_cast<const v4f*>(vbase + c * O_DIM);
                const v4f vlo = vp[0];
                const v4f vhi = vp[1];
                float part = 0.0f;
#pragma unroll
                for (int j = 0; j < 4; ++j) part += u[j] * vlo[j];
#pragma unroll
                for (int j = 0; j < 4; ++j) part += u[j + 4] * vhi[j];
                cij[c] = part + __shfl_xor(part, 16, 32);  // sum both o-halves
            }
            // ---- softmax over capsules; |logit| << 1 so no max-shift needed ----
            float ssum = 0.0f;
#pragma unroll
            for (int c = 0; c < C_CAPS; ++c) {
                const float e = __expf(cij[c]);
                cij[c] = e;
                ssum += e;
            }
            const float inv = 1.0f / ssum;

            // ---- weighted accumulate: scale B columns by cij, WMMA into s ----
#pragma unroll
            for (int c = 0; c < C_CAPS; ++c) {
                const float w = cij[c] * inv;
                const v2f wb0 = bfrag0 * w;
                const v2f wb1 = bfrag1 * w;
                const float* wa = wp + c * (O_DIM * I_DIM);
                const v2f a0 = *reinterpret_cast<const v2f*>(wa);
                const v2f a1 = *reinterpret_cast<const v2f*>(wa + 4);
                s_wacc[c] = __builtin_amdgcn_wmma_f32_16x16x4_f32(
                        false, a0, false, wb0, (short)0, s_wacc[c], false, false);
                s_wacc[c] = __builtin_amdgcn_wmma_f32_16x16x4_f32(
                        false, a1, false, wb1, (short)0, s_wacc[c], false, false);
            }
        }
        xp += I_DIM;        // next route: x stride 8 floats
        wp += W_RSTRIDE;    // next route: W stride 1280 floats
    }

    // cross-wave reduction of s_j via LDS float atomics (ds_add_f32)
#pragma unroll
    for (int c = 0; c < C_CAPS; ++c)
#pragma unroll
        for (int j = 0; j < 8; ++j)
            atomicAdd(&s_lds[(bl * C_CAPS + c) * O_DIM + 8 * hh + j],
                      s_wacc[c][j]);
    __syncthreads();

    // squash on wave 0: 160 (b,c) pairs over 32 lanes
    if (wave == 0) {
        for (int it = 0; it < 5; ++it) {
            const int idx = it * 32 + lane;       // = b*10 + c
            const int c = idx % C_CAPS;
            const int b = idx / C_CAPS;
            float sv[O_DIM];
            float nsq = 0.0f;
#pragma unroll
            for (int o = 0; o < O_DIM; ++o) {
                const float t = s_lds[idx * O_DIM + o] + bias[c * O_DIM + o];
                sv[o] = t;
                nsq += t * t;
            }
            const float nrm   = sqrtf(nsq);
            const float scale = nrm / (1.0f + nsq + 1e-8f);
            if constexpr (PHASE == 0) {
#pragma unroll
                for (int o = 0; o < O_DIM; ++o)
                    vdot_lds[idx * O_DIM + o] = scale * sv[o];       // v0
            } else if constexpr (PHASE == 1) {
#pragma unroll
                for (int o = 0; o < O_DIM; ++o)
                    vdot_lds[idx * O_DIM + o] += scale * sv[o];      // v0+v1
            } else {
#pragma unroll
                for (int o = 0; o < O_DIM; ++o)
                    out[(((size_t)(mb + b)) * C_CAPS + c) * O_DIM + o] =
                        scale * sv[o];                               // v2
            }
        }
    }
    __syncthreads();
}

__global__ __launch_bounds__(THREADS)
void capsule_routing_fused(const float* __restrict__ x,     // [512,1152,8]
                           const float* __restrict__ W,     // [1152,10,16,8]
                           const float* __restrict__ bias,  // [10,16]
                           float* __restrict__ out)         // [512,10,16]
{
    __shared__ float s_lds[S_ELEMS];     // cross-wave s_j reduction [b][c][o]
    __shared__ float vdot_lds[S_ELEMS];  // running v (p1: v0, p2: v0+v1)

    const int tid  = threadIdx.x;
    const int wave = tid >> 5;
    const int lane = tid & 31;
    const int bl   = lane & 15;   // lane row: batch col for B/D, o-row for A
    const int hh   = lane >> 4;   // lane half
    const int mb   = blockIdx.x * MTILE;

    phase_body<0>(x, W, bias, out, s_lds, vdot_lds, tid, wave, lane, bl, hh, mb);
    phase_body<1>(x, W, bias, out, s_lds, vdot_lds, tid, wave, lane, bl, hh, mb);
    phase_body<2>(x, W, bias, out, s_lds, vdot_lds, tid, wave, lane, bl, hh, mb);
}

extern "C" void kernel_launch(void* const* d_in, const int* in_sizes, int n_in,
                              void* d_out, int out_size, void* d_ws, size_t ws_size,
                              hipStream_t stream) {
    const float* x    = (const float*)d_in[0];
    const float* W    = (const float*)d_in[1];
    const float* bias = (const float*)d_in[2];
    float* out        = (float*)d_out;
    (void)in_sizes; (void)n_in; (void)out_size; (void)d_ws; (void)ws_size;

    dim3 grid(B_TOT / MTILE);   // 32 blocks
    dim3 block(THREADS);        // 8 wave32 waves
    capsule_routing_fused<<<grid, block, 0, stream>>>(x, W, bias, out);
}